// LogitsProcessor_9844065042736
// MI455X (gfx1250) — compile-verified
//
#include <hip/hip_runtime.h>
#include <math.h>

// Problem constants (match the reference exactly).
#define Bn 64
#define Dn 2048
#define Vn 128000
#define CHUNK 256                 // vocab columns per block (8 waves x 2 groups x 16)
#define NCHUNK (Vn / CHUNK)       // 500

typedef __attribute__((ext_vector_type(2))) float v2f;
typedef __attribute__((ext_vector_type(8))) float v8f;

#define WMMA_F32(A, B, C) \
    __builtin_amdgcn_wmma_f32_16x16x4_f32(false, (A), false, (B), (short)0, (C), false, false)

struct ChunkStat {
    float m;        // chunk max of scaled logits
    float s;        // sum exp(x - m) over chunk
    float amax_val; // chunk argmax value (scaled logit)
    int   amax_idx; // chunk argmax vocab index
    float gkey;     // chunk max of (scaled logit + gumbel)
    float gx;       // scaled logit at gumbel winner
    int   gidx;     // gumbel winner vocab index
    int   pad;
};

// ---------------------------------------------------------------------------
// Kernel 1: WMMA GEMM over a 64 x 256 logit tile + fused chunk-local softmax /
// argmax / gumbel-argmax statistics. One block per vocab chunk; each block
// computes ALL 64 batch rows so the 1.05 GB weight matrix streams from HBM
// exactly once (NT hint keeps it from thrashing L2; hs stays L2-resident).
// Each wave owns two 16-column groups -> 8 WMMAs per 6 loads per K-step, and
// the loop is software-pipelined (prefetch regs) so the pre-WMMA wait can
// leave the next step's 6 loads in flight instead of draining to zero.
// ---------------------------------------------------------------------------
__global__ __launch_bounds__(256) void logits_wmma_stats_kernel(
    const float* __restrict__ hs,      // [Bn, Dn]
    const float* __restrict__ wt,      // [Vn, Dn] row-major
    const float* __restrict__ temp,    // [Bn]
    const float* __restrict__ gum,     // [Bn, Vn]
    const int*   __restrict__ lmidx,   // [Bn] gather indices
    ChunkStat*   __restrict__ stats)   // [Bn, NCHUNK]
{
    __shared__ float xs[Bn][CHUNK + 4];            // pitch 260: rows r and r+8 land 32 banks apart
    __shared__ float pm[256], ps[256], pav[256], pgk[256], pgx[256];
    __shared__ int   pai[256], pgi[256];

    const int tid   = threadIdx.x;
    const int wv    = tid >> 5;        // wave id (wave32)
    const int lane  = tid & 31;
    const int hh    = lane >> 4;       // lane half selects K offset {0,2}
    const int ln    = lane & 15;       // M (for A) / N (for B)
    const int chunk = blockIdx.x;
    const int wcol0 = chunk * CHUNK + wv * 32;

    // A operand base pointers: 4 row-tiles of 16, gathered via lm_head_indices.
    // ISA 32-bit A 16x4 layout: lane holds float2 {K=2h, K=2h+1} of row (lane&15).
    const float* arow0 = hs + (long)lmidx[ 0 + ln] * Dn + 2 * hh;
    const float* arow1 = hs + (long)lmidx[16 + ln] * Dn + 2 * hh;
    const float* arow2 = hs + (long)lmidx[32 + ln] * Dn + 2 * hh;
    const float* arow3 = hs + (long)lmidx[48 + ln] * Dn + 2 * hh;
    // B operands (4x16, K x N): lane holds float2 {K=2h, K=2h+1} of column (lane&15);
    // two independent 16-column groups per wave.
    const float* brow0 = wt + (long)(wcol0 +      ln) * Dn + 2 * hh;
    const float* brow1 = wt + (long)(wcol0 + 16 + ln) * Dn + 2 * hh;

    v8f a00 = {}, a01 = {}, a02 = {}, a03 = {};   // column group 0, row tiles 0..3
    v8f a10 = {}, a11 = {}, a12 = {}, a13 = {};   // column group 1, row tiles 0..3

    // Software pipeline: preload K-step 0, prefetch K-step k+4 before consuming k.
    v2f b0 = __builtin_nontemporal_load((const v2f*)(brow0));
    v2f b1 = __builtin_nontemporal_load((const v2f*)(brow1));
    v2f f0 = *(const v2f*)(arow0);
    v2f f1 = *(const v2f*)(arow1);
    v2f f2 = *(const v2f*)(arow2);
    v2f f3 = *(const v2f*)(arow3);

#pragma unroll 2
    for (int k0 = 0; k0 < Dn - 4; k0 += 4) {
        const int kn = k0 + 4;
        // Issue next step's loads first so they overlap the WMMAs below.
        v2f nb0 = __builtin_nontemporal_load((const v2f*)(brow0 + kn));
        v2f nb1 = __builtin_nontemporal_load((const v2f*)(brow1 + kn));
        v2f nf0 = *(const v2f*)(arow0 + kn);
        v2f nf1 = *(const v2f*)(arow1 + kn);
        v2f nf2 = *(const v2f*)(arow2 + kn);
        v2f nf3 = *(const v2f*)(arow3 + kn);

        a00 = WMMA_F32(f0, b0, a00);
        a01 = WMMA_F32(f1, b0, a01);
        a02 = WMMA_F32(f2, b0, a02);
        a03 = WMMA_F32(f3, b0, a03);
        a10 = WMMA_F32(f0, b1, a10);
        a11 = WMMA_F32(f1, b1, a11);
        a12 = WMMA_F32(f2, b1, a12);
        a13 = WMMA_F32(f3, b1, a13);

        b0 = nb0; b1 = nb1;
        f0 = nf0; f1 = nf1; f2 = nf2; f3 = nf3;
    }
    // Epilogue: last K-step (k0 = Dn-4), operands already in registers.
    a00 = WMMA_F32(f0, b0, a00);
    a01 = WMMA_F32(f1, b0, a01);
    a02 = WMMA_F32(f2, b0, a02);
    a03 = WMMA_F32(f3, b0, a03);
    a10 = WMMA_F32(f0, b1, a10);
    a11 = WMMA_F32(f1, b1, a11);
    a12 = WMMA_F32(f2, b1, a12);
    a13 = WMMA_F32(f3, b1, a13);

    // C/D layout: vgpr j, lane l -> row = j + 8*(l>>4), col = l&15.
    const int c0 = wv * 32 + ln;
    const int c1 = c0 + 16;
#pragma unroll
    for (int j = 0; j < 8; ++j) {
        const int rj = j + 8 * hh;
        xs[ 0 + rj][c0] = a00[j];
        xs[16 + rj][c0] = a01[j];
        xs[32 + rj][c0] = a02[j];
        xs[48 + rj][c0] = a03[j];
        xs[ 0 + rj][c1] = a10[j];
        xs[16 + rj][c1] = a11[j];
        xs[32 + rj][c1] = a12[j];
        xs[48 + rj][c1] = a13[j];
    }
    __syncthreads();

    // Chunk-local statistics: 4 threads per row, each scans 64 columns.
    const int r = tid >> 2;   // 0..63
    const int q = tid & 3;    // 0..3
    const float invt = 1.0f / temp[r];
    const float* grow = gum + (long)r * Vn + (long)chunk * CHUNK;

    float m = -__builtin_inff(), s = 0.0f;
    float av = -__builtin_inff(); int ai = 0x7fffffff;
    float gk = -__builtin_inff(), gx = 0.0f; int gi = 0x7fffffff;
#pragma unroll 4
    for (int i = 0; i < 64; ++i) {
        const int c    = q * 64 + i;
        const float x  = xs[r][c] * invt;
        const int gcol = chunk * CHUNK + c;
        const float mn = fmaxf(m, x);
        s = s * __expf(m - mn) + __expf(x - mn);
        m = mn;
        if (x > av) { av = x; ai = gcol; }
        const float key = x + __builtin_nontemporal_load(grow + c);
        if (key > gk) { gk = key; gx = x; gi = gcol; }
    }

    pm[tid] = m; ps[tid] = s; pav[tid] = av; pai[tid] = ai;
    pgk[tid] = gk; pgx[tid] = gx; pgi[tid] = gi;
    __syncthreads();

    if (q == 0) {
#pragma unroll
        for (int o = 1; o < 4; ++o) {
            const int u = tid + o;
            const float m2 = pm[u], s2 = ps[u];
            const float mn = fmaxf(m, m2);
            s = s * __expf(m - mn) + s2 * __expf(m2 - mn);
            m = mn;
            if (pav[u] > av || (pav[u] == av && pai[u] < ai)) { av = pav[u]; ai = pai[u]; }
            if (pgk[u] > gk || (pgk[u] == gk && pgi[u] < gi)) { gk = pgk[u]; gx = pgx[u]; gi = pgi[u]; }
        }
        ChunkStat st;
        st.m = m; st.s = s; st.amax_val = av; st.amax_idx = ai;
        st.gkey = gk; st.gx = gx; st.gidx = gi; st.pad = 0;
        stats[(long)r * NCHUNK + chunk] = st;
    }
}

// ---------------------------------------------------------------------------
// Kernel 2: per-row merge of the 500 chunk stats (online-softmax merge +
// index-tie-break-low argmax merge), then finalize id + logprob.
// ---------------------------------------------------------------------------
__global__ __launch_bounds__(256) void finalize_sample_kernel(
    const ChunkStat* __restrict__ stats,
    const int*       __restrict__ gmask,   // [Bn] greedy flags
    float*           __restrict__ out)     // [0..63]=ids, [64..127]=logprobs
{
    __shared__ float pm[256], ps[256], pav[256], pgk[256], pgx[256];
    __shared__ int   pai[256], pgi[256];

    const int r   = blockIdx.x;
    const int tid = threadIdx.x;

    float m = -__builtin_inff(), s = 0.0f;
    float av = -__builtin_inff(); int ai = 0x7fffffff;
    float gk = -__builtin_inff(), gx = 0.0f; int gi = 0x7fffffff;

    for (int c = tid; c < NCHUNK; c += 256) {
        const ChunkStat st = stats[(long)r * NCHUNK + c];
        const float mn = fmaxf(m, st.m);
        s = s * __expf(m - mn) + st.s * __expf(st.m - mn);
        m = mn;
        if (st.amax_val > av || (st.amax_val == av && st.amax_idx < ai)) { av = st.amax_val; ai = st.amax_idx; }
        if (st.gkey     > gk || (st.gkey     == gk && st.gidx     < gi)) { gk = st.gkey; gx = st.gx; gi = st.gidx; }
    }

    pm[tid] = m; ps[tid] = s; pav[tid] = av; pai[tid] = ai;
    pgk[tid] = gk; pgx[tid] = gx; pgi[tid] = gi;
    __syncthreads();

    for (int off = 128; off > 0; off >>= 1) {
        if (tid < off) {
            const int u = tid + off;
            const float m1 = pm[tid], s1 = ps[tid], m2 = pm[u], s2 = ps[u];
            const float mn = fmaxf(m1, m2);
            ps[tid] = s1 * __expf(m1 - mn) + s2 * __expf(m2 - mn);
            pm[tid] = mn;
            if (pav[u] > pav[tid] || (pav[u] == pav[tid] && pai[u] < pai[tid])) {
                pav[tid] = pav[u]; pai[tid] = pai[u];
            }
            if (pgk[u] > pgk[tid] || (pgk[u] == pgk[tid] && pgi[u] < pgi[tid])) {
                pgk[tid] = pgk[u]; pgx[tid] = pgx[u]; pgi[tid] = pgi[u];
            }
        }
        __syncthreads();
    }

    if (tid == 0) {
        const bool greedy = (gmask[r] != 0);
        const float logZ  = pm[0] + logf(ps[0]);
        const int   id    = greedy ? pai[0] : pgi[0];
        const float xv    = greedy ? pav[0] : pgx[0];
        out[r]      = (float)id;       // token id (exact: id < 2^24)
        out[Bn + r] = xv - logZ;       // chosen log-probability
    }
}

extern "C" void kernel_launch(void* const* d_in, const int* in_sizes, int n_in,
                              void* d_out, int out_size, void* d_ws, size_t ws_size,
                              hipStream_t stream) {
    const float* hs     = (const float*)d_in[0];   // hidden_states [64,2048] f32
    const float* wt     = (const float*)d_in[1];   // weight [128000,2048] f32
    const float* temp   = (const float*)d_in[2];   // temperature [64] f32
    const float* gum    = (const float*)d_in[3];   // gumbel [64,128000] f32
    const int*   lmidx  = (const int*)d_in[4];     // lm_head_indices [64]
    const int*   gmask  = (const int*)d_in[5];     // greedy_mask [64]
    ChunkStat*   stats  = (ChunkStat*)d_ws;        // 64*500*32 B = 1 MB scratch
    (void)in_sizes; (void)n_in; (void)out_size; (void)ws_size;

    logits_wmma_stats_kernel<<<NCHUNK, 256, 0, stream>>>(hs, wt, temp, gum, lmidx, stats);
    finalize_sample_kernel<<<Bn, 256, 0, stream>>>(stats, gmask, (float*)d_out);
}